// GraphSageWithSampling_43568148251366
// MI455X (gfx1250) — compile-verified
//
#include <hip/hip_runtime.h>
#include <math.h>
#include <stdint.h>

typedef float v2f __attribute__((ext_vector_type(2)));
typedef float v8f __attribute__((ext_vector_type(8)));

__device__ __forceinline__ float lrelu(float x) { return x > 0.f ? x : 0.1f * x; }

// ---------------- utility kernels ----------------

__global__ void k_zero(float* __restrict__ p, int n) {
    int i = blockIdx.x * 256 + threadIdx.x;
    if (i < n) p[i] = 0.f;
}

// gather emb rows through node_ids -> dense [N, ED]
__global__ void k_gather(const float* __restrict__ emb, const int* __restrict__ ids,
                         float* __restrict__ out, int n, int d) {
    int i = blockIdx.x * 256 + threadIdx.x;
    if (i < n * d) {
        int r = i / d, c = i % d;
        out[i] = emb[(long)ids[r] * d + c];
    }
}

// in-degree (float) via atomics
__global__ void k_deg(const int* __restrict__ dst, float* __restrict__ deg, int nE) {
    int e = blockIdx.x * 256 + threadIdx.x;
    if (e < nE) atomicAdd(&deg[dst[e]], 1.0f);
}

// zero the aggregation half (cols 256..511) of hc [N,512]
__global__ void k_zero_agg(float* __restrict__ hc, int n /* = N*256 */) {
    int i = blockIdx.x * 256 + threadIdx.x;
    if (i < n) {
        int r = i >> 8, c = i & 255;
        hc[(long)r * 512 + 256 + c] = 0.f;
    }
}

// edge scatter: agg[dst] += h[src].  One wave per edge, lane covers 8 features.
// h (51MB) and agg (51MB) are both L2-resident (192MB L2) -> atomics resolve in L2.
__global__ void k_scatter(const float* __restrict__ h, const int* __restrict__ src,
                          const int* __restrict__ dst, float* __restrict__ hc, int nE) {
    int wave = threadIdx.x >> 5;
    int lane = threadIdx.x & 31;
    int e0 = blockIdx.x * 64;                 // 64 edges per block (8 waves x 8 passes)
    for (int p = 0; p < 8; ++p) {
        int e = e0 + p * 8 + wave;
        if (e < nE) {
            int s = src[e], d = dst[e];
            const float* hs = h + (long)s * 256;
            float* ag = hc + (long)d * 512 + 256;
#pragma unroll
            for (int j = 0; j < 8; ++j) {
                atomicAdd(&ag[lane + 32 * j], hs[lane + 32 * j]);
            }
        }
    }
}

// build concat input: hc[:,0:256]=h ; hc[:,256:512]=agg/max(deg,1)
__global__ void k_finalize(const float* __restrict__ h, const float* __restrict__ deg,
                           float* __restrict__ hc, int n /* = N*256 */) {
    int i = blockIdx.x * 256 + threadIdx.x;
    if (i < n) {
        int r = i >> 8, c = i & 255;
        float inv = 1.f / fmaxf(deg[r], 1.f);
        long b = (long)r * 512;
        hc[b + c] = h[i];
        hc[b + 256 + c] *= inv;
    }
}

// row-wise L2 normalization, one wave32 per row of 256
__global__ void k_l2norm(const float* __restrict__ in, float* __restrict__ out, int nRows) {
    int wave = threadIdx.x >> 5, lane = threadIdx.x & 31;
    int row = blockIdx.x * 8 + wave;
    if (row >= nRows) return;
    const float* x = in + (long)row * 256;
    float v[8];
    float s = 0.f;
#pragma unroll
    for (int j = 0; j < 8; ++j) { v[j] = x[lane + 32 * j]; s += v[j] * v[j]; }
#pragma unroll
    for (int m = 16; m >= 1; m >>= 1) s += __shfl_xor(s, m, 32);
    float inv = 1.f / fmaxf(sqrtf(s), 1e-6f);
    float* o = out + (long)row * 256;
#pragma unroll
    for (int j = 0; j < 8; ++j) o[lane + 32 * j] = v[j] * inv;
}

// ---------------- fp32 WMMA GEMM with async global->LDS tile staging ----------
// out[M,N] = act(A[M,K] @ W[N,K]^T + bias[N]) (+ residual[M,N])
// 256 threads = 8 waves; block tile 32(M) x 64(N); each wave owns one 16x16
// V_WMMA_F32_16X16X4_F32 tile; K chunked by 64 through padded LDS.
// Full tiles stream in via GLOBAL_LOAD_ASYNC_TO_LDS_B128 (no VGPR round trip,
// tracked by ASYNCcnt); edge tiles fall back to masked VALU loads.
#define BK   64
#define LPAD 4   // stride 68 floats = 272B: 16B-aligned rows, 68%64=4 -> conflict-free

__global__ void __launch_bounds__(256)
k_gemm(const float* __restrict__ A, const float* __restrict__ W,
       const float* __restrict__ bias, const float* __restrict__ residual,
       float* __restrict__ out, int M, int N, int K, int doLrelu) {
    __shared__ __align__(16) float As[32][BK + LPAD];
    __shared__ __align__(16) float Bs[64][BK + LPAD];

    int tid = threadIdx.x;
    int wave = tid >> 5, lane = tid & 31;
    int subRow = wave >> 2;            // 0..1 -> M offset 16*subRow
    int subCol = wave & 3;             // 0..3 -> N offset 16*subCol
    int rowBase = blockIdx.y * 32;
    int colBase = blockIdx.x * 64;
    int lrow = lane & 15;
    int hi = lane >> 4;                // lane half selects K pair {0,1} vs {2,3}

    v8f acc = {0.f, 0.f, 0.f, 0.f, 0.f, 0.f, 0.f, 0.f};

    for (int k0 = 0; k0 < K; k0 += BK) {
        bool fullK = (k0 + BK <= K);
        // ---- A tile 32 x 64 ----
        if (fullK && (rowBase + 32 <= M)) {
#pragma unroll
            for (int j = 0; j < 2; ++j) {          // 512 float4 slots / 256 threads
                int i = tid + 256 * j;
                int r = i >> 4, c4 = (i & 15) << 2;
                unsigned lds = (unsigned)(uintptr_t)&As[r][c4];
                unsigned long long ga =
                    (unsigned long long)(uintptr_t)(A + (size_t)(rowBase + r) * K + k0 + c4);
                asm volatile("global_load_async_to_lds_b128 %0, %1, off"
                             :: "v"(lds), "v"(ga) : "memory");
            }
        } else {
            for (int i = tid; i < 32 * BK; i += 256) {
                int r = i / BK, c = i % BK;
                int gr = rowBase + r, gc = k0 + c;
                As[r][c] = (gr < M && gc < K) ? A[(long)gr * K + gc] : 0.f;
            }
        }
        // ---- B tile 64 x 64 (rows of W = output cols) ----
        if (fullK && (colBase + 64 <= N)) {
#pragma unroll
            for (int j = 0; j < 4; ++j) {          // 1024 float4 slots / 256 threads
                int i = tid + 256 * j;
                int r = i >> 4, c4 = (i & 15) << 2;
                unsigned lds = (unsigned)(uintptr_t)&Bs[r][c4];
                unsigned long long ga =
                    (unsigned long long)(uintptr_t)(W + (size_t)(colBase + r) * K + k0 + c4);
                asm volatile("global_load_async_to_lds_b128 %0, %1, off"
                             :: "v"(lds), "v"(ga) : "memory");
            }
        } else {
            for (int i = tid; i < 64 * BK; i += 256) {
                int r = i / BK, c = i % BK;
                int gr = colBase + r, gc = k0 + c;
                Bs[r][c] = (gr < N && gc < K) ? W[(long)gr * K + gc] : 0.f;
            }
        }
        asm volatile("s_wait_asynccnt 0x0" ::: "memory");
        __syncthreads();

#pragma unroll
        for (int kk = 0; kk < BK; kk += 4) {
            // A 16x4 fragment: v0 = K{0|2}, v1 = K{1|3} by lane half (ISA 7.12.2)
            v2f a, b;
            a.x = As[16 * subRow + lrow][kk + 2 * hi];
            a.y = As[16 * subRow + lrow][kk + 2 * hi + 1];
            // B 4x16 fragment (B[k][n] = W[n][k]), mirrored layout
            b.x = Bs[16 * subCol + lrow][kk + 2 * hi];
            b.y = Bs[16 * subCol + lrow][kk + 2 * hi + 1];
            acc = __builtin_amdgcn_wmma_f32_16x16x4_f32(
                false, a, false, b, (short)0, acc, false, false);
        }
        __syncthreads();
    }

    // epilogue: C/D layout -> vgpr j = row j + 8*hi, col = lane&15
    int col = colBase + 16 * subCol + lrow;
#pragma unroll
    for (int j = 0; j < 8; ++j) {
        int row = rowBase + 16 * subRow + j + 8 * hi;
        if (row < M && col < N) {
            float v = acc[j] + bias[col];
            if (doLrelu) v = lrelu(v);
            if (residual) v += residual[(long)row * N + col];
            out[(long)row * N + col] = v;
        }
    }
}

// ---------------- host driver ----------------

static inline int cdiv(int a, int b) { return (a + b - 1) / b; }

extern "C" void kernel_launch(void* const* d_in, const int* in_sizes, int n_in,
                              void* d_out, int out_size, void* d_ws, size_t ws_size,
                              hipStream_t stream) {
    (void)in_sizes; (void)n_in; (void)out_size; (void)ws_size;
    const int N = 50000, NE = 1000000, F = 256, NC = 300, ED = 64, L = 3;

    const int*   node_ids = (const int*)d_in[0];
    const float* content  = (const float*)d_in[1];
    const int*   src      = (const int*)d_in[2];
    const int*   dst      = (const int*)d_in[3];
    const float* emb      = (const float*)d_in[4];
    const float* W_exp    = (const float*)d_in[5];
    const float* b_exp    = (const float*)d_in[6];
    const float* W_p1     = (const float*)d_in[7];
    const float* b_p1     = (const float*)d_in[8];
    const float* W_p2     = (const float*)d_in[9];
    const float* b_p2     = (const float*)d_in[10];
    const float* W_conv   = (const float*)d_in[11];
    const float* b_conv   = (const float*)d_in[12];
    const float* Wo1      = (const float*)d_in[13];
    const float* bo1      = (const float*)d_in[14];
    const float* Wo2      = (const float*)d_in[15];
    const float* bo2      = (const float*)d_in[16];

    // workspace layout (floats)
    float* ws   = (float*)d_ws;
    float* h    = ws;                                   // [N,256]
    float* hc   = h    + (size_t)N * F;                 // [N,512] (concat buffer)
    float* tmpA = hc   + (size_t)N * 2 * F;             // [N,300] / reused as hn [N,256]
    float* tmpB = tmpA + (size_t)N * NC;                // [N,256] / embg [N,64]
    float* deg  = tmpB + (size_t)N * F;                 // [N]

    dim3 blk(256);
    const int rowTiles = cdiv(N, 32);

    // 1) gather embedding rows
    k_gather<<<cdiv(N * ED, 256), blk, 0, stream>>>(emb, node_ids, tmpB, N, ED);

    // 2) h = lrelu(embg @ W_exp^T + b_exp)        [N,64] x [256,64]^T
    k_gemm<<<dim3(cdiv(F, 64), rowTiles), blk, 0, stream>>>(
        tmpB, W_exp, b_exp, nullptr, h, N, F, ED, 1);

    // 3) tmpA = lrelu(content @ W_p1^T + b_p1)    [N,300] x [300,300]^T
    k_gemm<<<dim3(cdiv(NC, 64), rowTiles), blk, 0, stream>>>(
        content, W_p1, b_p1, nullptr, tmpA, N, NC, NC, 1);

    // 4) h = lrelu(tmpA @ W_p2^T + b_p2) + h      [N,300] x [256,300]^T  (residual fused)
    k_gemm<<<dim3(cdiv(F, 64), rowTiles), blk, 0, stream>>>(
        tmpA, W_p2, b_p2, h, h, N, F, NC, 1);

    // 5) in-degree
    k_zero<<<cdiv(N, 256), blk, 0, stream>>>(deg, N);
    k_deg<<<cdiv(NE, 256), blk, 0, stream>>>(dst, deg, NE);

    for (int i = 0; i < L; ++i) {
        // 6) agg = segment_sum(h[src] -> dst); then concat+mean into hc
        k_zero_agg<<<cdiv(N * F, 256), blk, 0, stream>>>(hc, N * F);
        k_scatter<<<cdiv(NE, 64), blk, 0, stream>>>(h, src, dst, hc, NE);
        k_finalize<<<cdiv(N * F, 256), blk, 0, stream>>>(h, deg, hc, N * F);

        // 7) h_new = (lrelu?)(hc @ W_conv[i]^T + b_conv[i])   K = 512
        k_gemm<<<dim3(cdiv(F, 64), rowTiles), blk, 0, stream>>>(
            hc, W_conv + (size_t)i * F * 2 * F, b_conv + (size_t)i * F,
            nullptr, tmpB, N, F, 2 * F, (i < L - 1) ? 1 : 0);

        // 8) row L2 normalize
        float* nrmOut = (i < L - 1) ? tmpA : (float*)d_out;
        k_l2norm<<<cdiv(N, 8), blk, 0, stream>>>(tmpB, nrmOut, N);

        if (i < L - 1) {
            // 9) tmpB = lrelu(hn @ Wo1[i]^T + bo1[i])
            k_gemm<<<dim3(cdiv(F, 64), rowTiles), blk, 0, stream>>>(
                tmpA, Wo1 + (size_t)i * F * F, bo1 + (size_t)i * F,
                nullptr, tmpB, N, F, F, 1);
            // 10) h = tmpB @ Wo2[i]^T + bo2[i]
            k_gemm<<<dim3(cdiv(F, 64), rowTiles), blk, 0, stream>>>(
                tmpB, Wo2 + (size_t)i * F * F, bo2 + (size_t)i * F,
                nullptr, h, N, F, F, 0);
        }
    }
}